// PTSNetwork_46196668236356
// MI455X (gfx1250) — compile-verified
//
#include <hip/hip_runtime.h>
#include <stdint.h>

#define VOCAB 50257
#define TOPK  10
#define NN    5
#define BLOCK 1024
#define NV4   (VOCAB / 4)                    // 12564 float4 chunks = 50256 floats
#define NCHUNK 13                            // ceil(NV4 / BLOCK)
#define ROW_PAD 50260                        // row region in floats (16B aligned)
#define LIST_OFF ROW_PAD                     // per-thread top-10 lists: 1024*10 floats
#define MISC_OFF (LIST_OFF + BLOCK * TOPK)   // 60500
#define SMEM_FLOATS (MISC_OFF + 80)          // 60580 floats = 242,320 B (< 320 KB LDS)
#define NEG_INF (-3.402823466e38f)
// misc layout: [0..49] W1 cache, [56..60] h1, [64..68] h2, [72] invT

typedef __attribute__((ext_vector_type(2))) float v2f;
typedef __attribute__((ext_vector_type(8))) float v8f;

__device__ __forceinline__ uint32_t lds_addr32(const void* p) {
  // generic shared pointer: addr[31:0] is the LDS byte address
  return (uint32_t)(uintptr_t)p;
}

__device__ __forceinline__ void topk_insert(float (&t)[TOPK], float v) {
  if (v > t[TOPK - 1]) {
    t[TOPK - 1] = v;
#pragma unroll
    for (int i = TOPK - 1; i >= 1; --i) {   // single bubble pass re-sorts
      float a = t[i - 1], b = t[i];
      t[i - 1] = fmaxf(a, b);
      t[i]     = fminf(a, b);
    }
  }
}

extern "C" __global__ void __launch_bounds__(BLOCK)
pts_softmax_kernel(const float* __restrict__ logits,
                   const float* __restrict__ W1, const float* __restrict__ b1,
                   const float* __restrict__ W2, const float* __restrict__ b2,
                   const float* __restrict__ W3, const float* __restrict__ b3,
                   float* __restrict__ out)
{
  extern __shared__ float smem[];
  float* row   = smem;             // [0, 50260)  the staged logits row
  float* lists = smem + LIST_OFF;  // [50260, 60500) merge lists / sum partials
  float* misc  = smem + MISC_OFF;  // [60500, 60580) W1 cache, h1, h2, invT

  const int tid = threadIdx.x;
  const size_t rowoff = (size_t)blockIdx.x * (size_t)VOCAB;
  const float* src = logits + rowoff;

  // ---------- Phase 0a: issue ALL async DMA chunks up front (13 per wave) ----------
#pragma unroll
  for (int k = 0; k < NCHUNK; ++k) {
    int idx = k * BLOCK + tid;
    if (idx > NV4 - 1) idx = NV4 - 1;        // clamp: duplicate same-data write, benign
    uint32_t l = lds_addr32(&row[idx * 4]);
    uint64_t g = (uint64_t)(uintptr_t)(src + idx * 4);
    asm volatile("global_load_async_to_lds_b128 %0, %1, off"
                 :: "v"(l), "v"(g) : "memory");
  }
  // stage W1 into LDS (overlaps with DMA), tail element via regular load
  if (tid < TOPK * NN) misc[tid] = W1[tid];
  float tailv = NEG_INF;
  if (tid == 0) { tailv = src[VOCAB - 1]; row[VOCAB - 1] = tailv; }

  // ---------- Phase 1: top-k pipelined under the DMA (in-order completion) ----------
  float t[TOPK];
#pragma unroll
  for (int i = 0; i < TOPK; ++i) t[i] = NEG_INF;
#pragma unroll
  for (int k = 0; k < NCHUNK; ++k) {
    // chunks complete in order per wave: asynccnt <= 12-k  =>  chunks 0..k landed
    asm volatile("s_wait_asynccnt %0" :: "n"(NCHUNK - 1 - k) : "memory");
    int idx = k * BLOCK + tid;
    if (idx < NV4) {                         // guard processing (not issue)
      float4 vv = *reinterpret_cast<const float4*>(&row[idx * 4]);
      topk_insert(t, vv.x);
      topk_insert(t, vv.y);
      topk_insert(t, vv.z);
      topk_insert(t, vv.w);
    }
  }
  if (tid == 0) topk_insert(t, tailv);       // 50257th element
#pragma unroll
  for (int i = 0; i < TOPK; ++i) lists[tid * TOPK + i] = t[i];
  __syncthreads();

  // ---------- Phase 2: barrier tree of bitonic 10+10 -> top-10 merges ----------
  for (int s = BLOCK / 2; s >= 1; s >>= 1) {
    if (tid < s) {
      float m[32];
#pragma unroll
      for (int i = 0; i < TOPK; ++i) m[i] = lists[tid * TOPK + i];       // descending
#pragma unroll
      for (int i = TOPK; i < 22; ++i) m[i] = NEG_INF;                    // valley
#pragma unroll
      for (int i = 0; i < TOPK; ++i) m[31 - i] = lists[(tid + s) * TOPK + i]; // ascending
      // bitonic merge (desc) of length-32 bitonic sequence, all-static indices
#pragma unroll
      for (int st = 16; st >= 1; st >>= 1) {
#pragma unroll
        for (int i = 0; i < 32; ++i) {
          if ((i & st) == 0) {
            float a = m[i], b = m[i + st];
            m[i]      = fmaxf(a, b);
            m[i + st] = fminf(a, b);
          }
        }
      }
#pragma unroll
      for (int i = 0; i < TOPK; ++i) lists[tid * TOPK + i] = m[i];
    }
    __syncthreads();
  }

  // all threads grab the row max (= top-1) before lists gets reused
  const float rmax = lists[0];

  // ---------- Phase 3: MLP. Layer 1 via V_WMMA_F32_16X16X4_F32 (wave 0) ----------
  if (tid < 32) {            // full wave -> EXEC all ones for WMMA
    const int lane = tid;
    const int col  = lane & 15;                         // B: N index
    const int colc = (col < NN) ? col : (NN - 1);
    const float colm = (col < NN) ? 1.f : 0.f;
    v8f c = {};              // C/D accumulator, zero-init
#pragma unroll
    for (int j = 0; j < 3; ++j) {          // K = 12 (>= 10), chained accumulation
      const int kb = 4 * j + ((lane & 16) ? 2 : 0);     // A: lanes 0-15 K0/K1, 16-31 K2/K3
      const int k0 = (kb     < TOPK) ? kb     : (TOPK - 1);
      const int k1 = (kb + 1 < TOPK) ? kb + 1 : (TOPK - 1);
      const float m0 = (kb     < TOPK) ? 1.f : 0.f;
      const float m1 = (kb + 1 < TOPK) ? 1.f : 0.f;
      // branch-free operand build: clamped LDS reads x 0/1 masks (no exec juggling)
      v2f A  = { lists[k0] * m0,
                 lists[k1] * m1 };
      v2f Bv = { misc[k0 * NN + colc] * (m0 * colm),
                 misc[k1 * NN + colc] * (m1 * colm) };
      c = __builtin_amdgcn_wmma_f32_16x16x4_f32(false, A, false, Bv,
                                                (short)0, c, false, false);
    }
    // C layout: VGPR0, lanes 0-15 => (M=0, N=lane): h1[n] on lane n
    if (lane < NN) misc[56 + lane] = fmaxf(c[0] + b1[lane], 0.f);
    if (lane < NN) {                       // layer 2: 5x5, scalar per lane
      float acc = b2[lane];
#pragma unroll
      for (int k = 0; k < NN; ++k) acc += misc[56 + k] * W2[k * NN + lane];
      misc[64 + lane] = fmaxf(acc, 0.f);
    }
    if (lane == 0) {                       // layer 3: 5x1 -> temperature
      float acc = b3[0];
#pragma unroll
      for (int k = 0; k < NN; ++k) acc += misc[64 + k] * W3[k];
      float T = fminf(fmaxf(fabsf(acc), 1e-12f), 1e12f);
      misc[72] = 1.0f / T;
    }
  }
  __syncthreads();
  const float invT = misc[72];

  // ---------- Phase 4: softmax denominator from LDS ----------
  float local = 0.f;
  for (int j = tid; j < VOCAB; j += BLOCK)
    local += __expf((row[j] - rmax) * invT);
  lists[tid] = local;                      // lists reused as reduction scratch
  __syncthreads();
  for (int s = BLOCK / 2; s > 0; s >>= 1) {
    if (tid < s) lists[tid] += lists[tid + s];
    __syncthreads();
  }
  const float invS = 1.0f / lists[0];

  // ---------- Phase 5: write probabilities (coalesced) ----------
  float* dst = out + rowoff;
  for (int j = tid; j < VOCAB; j += BLOCK)
    dst[j] = __expf((row[j] - rmax) * invT) * invS;
}

extern "C" void kernel_launch(void* const* d_in, const int* in_sizes, int n_in,
                              void* d_out, int out_size, void* d_ws, size_t ws_size,
                              hipStream_t stream) {
  const float* logits = (const float*)d_in[0];
  const float* W1 = (const float*)d_in[1];
  const float* b1 = (const float*)d_in[2];
  const float* W2 = (const float*)d_in[3];
  const float* b2 = (const float*)d_in[4];
  const float* W3 = (const float*)d_in[5];
  const float* b3 = (const float*)d_in[6];
  float* out = (float*)d_out;

  const int rows = in_sizes[0] / VOCAB;          // 2048
  const size_t shmem = (size_t)SMEM_FLOATS * sizeof(float);  // ~242 KB dynamic LDS
  (void)hipFuncSetAttribute((const void*)pts_softmax_kernel,
                            hipFuncAttributeMaxDynamicSharedMemorySize, (int)shmem);
  pts_softmax_kernel<<<rows, BLOCK, shmem, stream>>>(logits, W1, b1, W2, b2, W3, b3, out);
}